// LSTM_83013127897716
// MI455X (gfx1250) — compile-verified
//
#include <hip/hip_runtime.h>
#include <cmath>

// LSTM single step, L=3 stacked layers, H=2048, batch=1.
// Memory-bound fp32 GEMV through V_WMMA_F32_16X16X4_F32 (exact fp32 math),
// fed by 128-bit global loads (shuffle-free K-permuted tiling).

typedef __attribute__((ext_vector_type(2))) float v2f;
typedef __attribute__((ext_vector_type(4))) float v4f;
typedef __attribute__((ext_vector_type(8))) float v8f;

#define H      2048
#define R      (4 * H)          // 8192 gate rows per layer
#define WAVES  8                // waves per workgroup (K-split)
#define TPB    (WAVES * 32)     // 256 threads, wave32
#define KSLICE (H / WAVES)      // 256 K per wave per matrix

// One workgroup computes a 16-row tile of gates = Wi@x + Wh@h + bi + bh.
// Wave w covers K in [w*KSLICE, (w+1)*KSLICE) of BOTH matrices.
// Per 8 K-values: one global_load_b128 (weights) + one ds_load_b128 (vector)
// feed two WMMAs whose K-sets are {k,k+1,k+4,k+5} and {k+2,k+3,k+6,k+7}
// (legal since WMMA accumulates a sum over K; A and B use matching per-lane
// K ordering, so no cross-lane shuffle is required).
__global__ __launch_bounds__(TPB) void lstm_gemv_wmma(
    const float* __restrict__ Wi, const float* __restrict__ Wh,
    const float* __restrict__ xv, const float* __restrict__ hv,
    const float* __restrict__ bi, const float* __restrict__ bh,
    float* __restrict__ gates)
{
    __shared__ v4f sx4[H / 4];             // 16B-aligned staging for x
    __shared__ v4f sh4[H / 4];             // 16B-aligned staging for h
    __shared__ float part[WAVES][16];

    const int tid = threadIdx.x;

    // Stage x and h vectors into LDS with coalesced 128-bit loads.
    for (int i = tid; i < H / 4; i += TPB) {
        sx4[i] = ((const v4f*)xv)[i];
        sh4[i] = ((const v4f*)hv)[i];
    }
    __syncthreads();

    const float* sx = (const float*)sx4;
    const float* sh = (const float*)sh4;

    const int wave    = tid >> 5;
    const int lane    = tid & 31;
    const int rowBase = blockIdx.x * 16;
    const int m       = lane & 15;          // tile row owned by this lane (A side)
    const int hb      = (lane >> 4) << 2;   // K half-select: 0 (lanes 0-15) or 4
    const int k0      = wave * KSLICE;

    // A 16x4 fp32 layout (ISA 7.12.2): lanes 0-15 supply (K0,K1) of row m,
    // lanes 16-31 supply (K2,K3). B 4x16 fp32: lower lane half = K0/K1,
    // upper half = K2/K3; the vector is broadcast across all 16 columns so
    // every D column holds the same partial dot product.
    const float* __restrict__ wrow_i = Wi + (size_t)(rowBase + m) * H;
    const float* __restrict__ wrow_h = Wh + (size_t)(rowBase + m) * H;

    v8f acc = {};

#pragma unroll 4
    for (int k = k0; k < k0 + KSLICE; k += 8) {
        v4f q  = *(const v4f*)(wrow_i + k + hb);   // global_load_b128
        v4f xb = *(const v4f*)(sx + k + hb);       // ds_load_b128
        v2f a0 = __builtin_shufflevector(q, q, 0, 1);
        v2f b0 = __builtin_shufflevector(xb, xb, 0, 1);
        acc = __builtin_amdgcn_wmma_f32_16x16x4_f32(
                  false, a0, false, b0, (short)0, acc, false, false);
        v2f a1 = __builtin_shufflevector(q, q, 2, 3);
        v2f b1 = __builtin_shufflevector(xb, xb, 2, 3);
        acc = __builtin_amdgcn_wmma_f32_16x16x4_f32(
                  false, a1, false, b1, (short)0, acc, false, false);
    }
#pragma unroll 4
    for (int k = k0; k < k0 + KSLICE; k += 8) {
        v4f q  = *(const v4f*)(wrow_h + k + hb);
        v4f xb = *(const v4f*)(sh + k + hb);
        v2f a0 = __builtin_shufflevector(q, q, 0, 1);
        v2f b0 = __builtin_shufflevector(xb, xb, 0, 1);
        acc = __builtin_amdgcn_wmma_f32_16x16x4_f32(
                  false, a0, false, b0, (short)0, acc, false, false);
        v2f a1 = __builtin_shufflevector(q, q, 2, 3);
        v2f b1 = __builtin_shufflevector(xb, xb, 2, 3);
        acc = __builtin_amdgcn_wmma_f32_16x16x4_f32(
                  false, a1, false, b1, (short)0, acc, false, false);
    }

    // D layout (32-bit C/D 16x16): VGPR v = row v on lanes 0-15, row v+8 on
    // lanes 16-31; all columns equal, so lane 0 / lane 16 hold the answers.
    if (lane == 0) {
#pragma unroll
        for (int v = 0; v < 8; ++v) part[wave][v] = acc[v];
    } else if (lane == 16) {
#pragma unroll
        for (int v = 0; v < 8; ++v) part[wave][8 + v] = acc[v];
    }
    __syncthreads();

    // Deterministic fixed-order reduction across the 8 K-split waves + biases.
    if (tid < 16) {
        float s = bi[rowBase + tid] + bh[rowBase + tid];
#pragma unroll
        for (int w = 0; w < WAVES; ++w) s += part[w][tid];
        gates[rowBase + tid] = s;
    }
}

// Elementwise LSTM cell update: gate order i, f, g, o.
__global__ __launch_bounds__(256) void lstm_pointwise(
    const float* __restrict__ gates, const float* __restrict__ c0,
    float* __restrict__ hout)
{
    int i = blockIdx.x * 256 + threadIdx.x;
    if (i >= H) return;
    float gi = gates[i];
    float gf = gates[H + i];
    float gg = gates[2 * H + i];
    float go = gates[3 * H + i];
    float si = 1.0f / (1.0f + expf(-gi));
    float sf = 1.0f / (1.0f + expf(-gf));
    float tg = tanhf(gg);
    float so = 1.0f / (1.0f + expf(-go));
    float c  = sf * c0[i] + si * tg;
    hout[i]  = so * tanhf(c);
}

extern "C" void kernel_launch(void* const* d_in, const int* in_sizes, int n_in,
                              void* d_out, int out_size, void* d_ws, size_t ws_size,
                              hipStream_t stream) {
    (void)in_sizes; (void)n_in; (void)out_size; (void)ws_size;
    const float* x   = (const float*)d_in[0];   // [H]
    const float* Wih = (const float*)d_in[1];   // [3, 4H, H]
    const float* Whh = (const float*)d_in[2];   // [3, 4H, H]
    const float* bih = (const float*)d_in[3];   // [3, 4H]
    const float* bhh = (const float*)d_in[4];   // [3, 4H]
    const float* h0  = (const float*)d_in[5];   // [3, H]
    const float* c0  = (const float*)d_in[6];   // [3, H]
    float* out = (float*)d_out;                 // [H]

    float* gates = (float*)d_ws;                // R floats
    float* hbuf  = gates + R;                   // H floats

    const float* cur_x = x;
    for (int l = 0; l < 3; ++l) {
        const float* Wi = Wih + (size_t)l * R * H;
        const float* Wh = Whh + (size_t)l * R * H;
        lstm_gemv_wmma<<<R / 16, TPB, 0, stream>>>(
            Wi, Wh, cur_x, h0 + (size_t)l * H, bih + (size_t)l * R,
            bhh + (size_t)l * R, gates);
        float* ho = (l == 2) ? out : hbuf;
        lstm_pointwise<<<H / 256, 256, 0, stream>>>(gates, c0 + (size_t)l * H, ho);
        cur_x = ho;
    }
}